// Scene_55551107006644
// MI455X (gfx1250) — compile-verified
//
#include <hip/hip_runtime.h>
#include <stdint.h>

#define CC 3
#define HH 4096
#define WW 4096
#define PHH 64
#define PWW 64
#define NSRC 1000

// Toggle the gfx1250 async global->LDS staging path.
#define USE_ASYNC_LDS 1

__device__ __forceinline__ void wait_asynccnt0() {
#if defined(__has_builtin) && __has_builtin(__builtin_amdgcn_s_wait_asynccnt)
    __builtin_amdgcn_s_wait_asynccnt(0);
#else
    asm volatile("s_wait_asynccnt 0" ::: "memory");
#endif
}

// ---------------------------------------------------------------------------
// Kernel 1: zero the 201 MB frame with B128 stores (re-run every launch:
// harness poisons/reuses d_out across graph replays).
// ---------------------------------------------------------------------------
__global__ __launch_bounds__(256) void zero_frame_kernel(float4* __restrict__ out,
                                                         int n4) {
    int i = blockIdx.x * blockDim.x + threadIdx.x;
    if (i < n4) {
        out[i] = make_float4(0.f, 0.f, 0.f, 0.f);
    }
}

// ---------------------------------------------------------------------------
// Kernel 2: one block per (source, channel) 64x64 tile.
//   - stage 16 KB tile into LDS with async b128 copies (ASYNCcnt path)
//   - each of 256 threads then scatters 4 float4 = 16 floats with f32 atomics
// ---------------------------------------------------------------------------
__global__ __launch_bounds__(256) void scatter_add_kernel(
    const float* __restrict__ patches,
    const int*   __restrict__ origins,
    float*       __restrict__ frame) {

    __shared__ float tile[PHH * PWW];  // 16 KB

    const int blk = blockIdx.x;        // 0 .. NSRC*CC-1
    const int src = blk / CC;
    const int ch  = blk - src * CC;
    const int t   = threadIdx.x;       // 0..255

    const float* gsrc = patches + ((size_t)src * CC + ch) * (PHH * PWW);

#if USE_ASYNC_LDS
    // 1024 b128 chunks of 16 B cover the 16 KB tile; 4 per thread.
    #pragma unroll
    for (int k = 0; k < 4; ++k) {
        const int chunk = t + k * 256;
        const float* gp = gsrc + chunk * 4;                       // 16B aligned
        // Low 32 bits of a generic LDS pointer are the LDS byte offset.
        unsigned lds_off = (unsigned)(uintptr_t)(tile + chunk * 4);
        asm volatile("global_load_async_to_lds_b128 %0, %1, off"
                     :
                     : "v"(lds_off), "v"(gp)
                     : "memory");
    }
    wait_asynccnt0();   // this wave's async copies landed in LDS
    __syncthreads();    // all waves' copies landed
#else
    #pragma unroll
    for (int k = 0; k < 4; ++k) {
        const int chunk = t + k * 256;
        ((float4*)tile)[chunk] = ((const float4*)gsrc)[chunk];
    }
    __syncthreads();
#endif

    const int y0 = origins[src * 2 + 0];
    const int x0 = origins[src * 2 + 1];
    float* fbase = frame + (size_t)ch * HH * WW;

    // chunk -> (py, px4): 16 chunks per 64-float row
    #pragma unroll
    for (int k = 0; k < 4; ++k) {
        const int chunk = t + k * 256;
        const int py  = chunk >> 4;
        const int px4 = (chunk & 15) << 2;
        const float4 v = ((const float4*)tile)[chunk];
        float* dst = fbase + (size_t)(y0 + py) * WW + (x0 + px4);
        // x0 has arbitrary 4B alignment -> per-float f32 atomics (overlaps
        // between patches require atomicity; L2 atomic units do the RMW).
        atomicAdd(dst + 0, v.x);
        atomicAdd(dst + 1, v.y);
        atomicAdd(dst + 2, v.z);
        atomicAdd(dst + 3, v.w);
    }
}

// ---------------------------------------------------------------------------
extern "C" void kernel_launch(void* const* d_in, const int* in_sizes, int n_in,
                              void* d_out, int out_size, void* d_ws, size_t ws_size,
                              hipStream_t stream) {
    const float* patches = (const float*)d_in[0];   // [1000,3,64,64] f32
    const int*   origins = (const int*)d_in[1];     // [1000,2] i32
    float*       frame   = (float*)d_out;           // [3,4096,4096] f32

    const int n4 = out_size / 4;                    // 12,582,912 float4s
    zero_frame_kernel<<<(n4 + 255) / 256, 256, 0, stream>>>((float4*)frame, n4);

    scatter_add_kernel<<<NSRC * CC, 256, 0, stream>>>(patches, origins, frame);
}